// Encoder_89962384982587
// MI455X (gfx1250) — compile-verified
//
#include <hip/hip_runtime.h>
#include <hip/hip_bf16.h>
#include <math.h>

// Problem constants (from reference)
#define NCELL 2
#define CC    128     // n_feat
#define HH    129     // RES
#define WW    129     // RES

typedef float v4f __attribute__((ext_vector_type(4)));
typedef int   i32x4 __attribute__((ext_vector_type(4)));

// Explicit LDS (addrspace 3) volatile view: forces real ds_load_b32 that
// cannot be folded away (the async-DMA write isn't visible to alias analysis).
typedef volatile const __attribute__((address_space(3))) float* lds_cvfp;

#if defined(__gfx1250__) && __has_builtin(__builtin_amdgcn_global_load_async_to_lds_b128)
#define HAVE_ASYNC_LDS 1
#endif

// ---------------------------------------------------------------------------
// Stage 1: depthwise 3x3 conv (SAME, cross-correlation) + tanh, transposed
// output layout (N, H, W, C) so stage 2 gathers are channel-contiguous.
//   F : (N, C, H, W)   Wg : (C, 1, 3, 3)   A : (N, H, W, C)
// Block = one (n, y, x-chunk); 128 threads = channels; sliding 3-col window.
// ---------------------------------------------------------------------------
#define XCHUNK 16
#define NXCHUNK ((WW + XCHUNK - 1) / XCHUNK)   // 9

__global__ __launch_bounds__(128) void conv_tanh_tr_kernel(
    const float* __restrict__ F, const float* __restrict__ Wg,
    float* __restrict__ A) {
  const int b     = blockIdx.x;
  const int chunk = b % NXCHUNK;
  const int rem   = b / NXCHUNK;
  const int y     = rem % HH;
  const int n     = rem / HH;
  const int c     = threadIdx.x;          // channel, 0..127

  // per-channel 3x3 filter
  float w0 = Wg[c * 9 + 0], w1 = Wg[c * 9 + 1], w2 = Wg[c * 9 + 2];
  float w3 = Wg[c * 9 + 3], w4 = Wg[c * 9 + 4], w5 = Wg[c * 9 + 5];
  float w6 = Wg[c * 9 + 6], w7 = Wg[c * 9 + 7], w8 = Wg[c * 9 + 8];

  const float* base = F + ((size_t)(n * CC + c)) * (HH * WW);
  const bool t_ok = (y > 0);
  const bool b_ok = (y < HH - 1);
  const float* rt = base + (size_t)(y - 1) * WW;
  const float* rm = base + (size_t)y * WW;
  const float* rb = base + (size_t)(y + 1) * WW;

  float* outp = A + (((size_t)n * HH + y) * WW) * CC + c;

  const int x0 = chunk * XCHUNK;
  const int x1 = (x0 + XCHUNK < WW) ? (x0 + XCHUNK) : WW;

  auto ldcol = [&](int x, float& r0, float& r1, float& r2) {
    if (x < 0 || x >= WW) { r0 = 0.f; r1 = 0.f; r2 = 0.f; }
    else {
      r0 = t_ok ? rt[x] : 0.f;
      r1 = rm[x];
      r2 = b_ok ? rb[x] : 0.f;
    }
  };

  float a0, a1, a2, c0, c1, c2, e0, e1, e2;   // cols x-1, x, x+1 (rows y-1,y,y+1)
  ldcol(x0 - 1, a0, a1, a2);
  ldcol(x0,     c0, c1, c2);

  for (int x = x0; x < x1; ++x) {
    ldcol(x + 1, e0, e1, e2);
    float acc = w0 * a0 + w1 * c0 + w2 * e0
              + w3 * a1 + w4 * c1 + w5 * e1
              + w6 * a2 + w7 * c2 + w8 * e2;
    outp[(size_t)x * CC] = tanhf(acc);       // stays dirty/resident in L2
    a0 = c0; a1 = c1; a2 = c2;
    c0 = e0; c1 = e1; c2 = e2;
  }
}

// ---------------------------------------------------------------------------
// Stage 2: cosine-weighted bilinear sample from A(N,H,W,C), sum over cells.
// 32 points per 256-thread block (8 waves x 4 points). Coordinates are
// staged block-wide into LDS with GLOBAL_LOAD_ASYNC_TO_LDS_B128 (ASYNCcnt),
// then broadcast-read per wave through an AS(3) volatile pointer (real
// ds_load_b32). Each lane owns 4 channels as a float4 -> every corner
// gather is one global_load_b128 and the full wave pulls a coalesced 512B
// burst from the L2-resident alpha grid. Output rows are written with
// non-temporal b128 stores so the 134MB stream doesn't evict the 17MB
// alpha grid from L2.
// ---------------------------------------------------------------------------
#define PTS_PER_WAVE  4
#define PTS_PER_BLOCK 32            // 8 waves * 4 points

__global__ __launch_bounds__(256) void sample_cosine_kernel(
    const float* __restrict__ px, const float* __restrict__ py,
    const float* __restrict__ A, float* __restrict__ out, int P) {
  __shared__ float sx[PTS_PER_BLOCK];
  __shared__ float sy[PTS_PER_BLOCK];

  const int t       = threadIdx.x;
  const int w       = t >> 5;        // wave in block, 0..7
  const int lane    = t & 31;
  const int blockP0 = blockIdx.x * PTS_PER_BLOCK;
  const bool full   = (blockP0 + PTS_PER_BLOCK) <= P;

  if (full) {
#ifdef HAVE_ASYNC_LDS
    // Wave 0 issues 16 async b128 copies (256B of coords) global -> LDS.
    if (t < 16) {
      const float* gsrc = (t < 8) ? (px + blockP0 + (t & 7) * 4)
                                  : (py + blockP0 + (t & 7) * 4);
      float*       ldst = (t < 8) ? (sx + (t & 7) * 4)
                                  : (sy + (t & 7) * 4);
      __builtin_amdgcn_global_load_async_to_lds_b128(
          (__attribute__((address_space(1))) i32x4*)gsrc,
          (__attribute__((address_space(3))) i32x4*)ldst,
          /*offset=*/0, /*cpol=*/0);
    }
    __builtin_amdgcn_s_wait_asynccnt(0);   // issuing wave: LDS data landed
    asm volatile("" ::: "memory");          // don't let reads move/fold
    __syncthreads();                        // publish to all waves
#else
    if (t < PTS_PER_BLOCK) {
      sx[t] = px[blockP0 + t];
      sy[t] = py[blockP0 + t];
    }
    asm volatile("" ::: "memory");
    __syncthreads();
#endif
  } else {
    // tail block: prefetch coordinate lines, load directly below
    __builtin_prefetch(px + blockP0, 0, 3);
    __builtin_prefetch(py + blockP0, 0, 3);
  }

  // AS(3) volatile views of the staged coordinates -> ds_load_b32 broadcast
  lds_cvfp vsx = (lds_cvfp)sx;
  lds_cvfp vsy = (lds_cvfp)sy;

  const float PI_F = 3.14159265358979323846f;

#pragma unroll
  for (int u = 0; u < PTS_PER_WAVE; ++u) {
    const int li = w * PTS_PER_WAVE + u;   // 0..31 within block
    const int p  = blockP0 + li;
    if (p >= P) break;

    float fx, fy;
    if (full) { fx = vsx[li]; fy = vsy[li]; }   // LDS broadcast read
    else      { fx = px[p];   fy = py[p];   }

    v4f acc = {0.f, 0.f, 0.f, 0.f};

#pragma unroll
    for (int n = 0; n < NCELL; ++n) {
      const float off = 0.5f * (float)n;       // linspace(0, 1-1/N, N)
      // xs = x-1 (domain [0,2]); ix = (xs+1)/2*(W-2)+off = x*63.5+off
      // ys = 2y-1;               iy = (ys+1)/2*(H-2)+off = y*127 +off
      const float ix = fx * 63.5f + off;
      const float iy = fy * 127.0f + off;
      const float ixl = floorf(ix);
      const float iyt = floorf(iy);

      // cosine step weights
      const float dxr = 0.5f * (1.0f - cosf(PI_F * (ixl + 1.0f - ix)));
      const float dyb = 0.5f * (1.0f - cosf(PI_F * (iyt + 1.0f - iy)));
      const float dxl = 1.0f - dxr;
      const float dyt = 1.0f - dyb;
      const float w_nw = dxr * dyb;   // (iyt, ixl)
      const float w_ne = dxl * dyb;   // (iyt, ixr)
      const float w_sw = dxr * dyt;   // (iyb, ixl)
      const float w_se = dxl * dyt;   // (iyb, ixr)

      int xl = min(max((int)ixl,     0), WW - 1);
      int xr = min(max((int)ixl + 1, 0), WW - 1);
      int yt = min(max((int)iyt,     0), HH - 1);
      int yb = min(max((int)iyt + 1, 0), HH - 1);

      const float* cellb = A + ((size_t)n * HH * WW) * CC;
      const v4f* row_t = (const v4f*)(cellb + ((size_t)yt * WW) * CC);
      const v4f* row_b = (const v4f*)(cellb + ((size_t)yb * WW) * CC);

      v4f v;
      v = row_t[(size_t)xl * 32 + lane]; acc += v * w_nw;
      v = row_t[(size_t)xr * 32 + lane]; acc += v * w_ne;
      v = row_b[(size_t)xl * 32 + lane]; acc += v * w_sw;
      v = row_b[(size_t)xr * 32 + lane]; acc += v * w_se;
    }

    // coalesced 512B/point non-temporal store (TH=NT): bypass-friendly stream
    v4f* orow = (v4f*)(out) + (size_t)p * 32 + lane;
    __builtin_nontemporal_store(acc, orow);
  }
}

// ---------------------------------------------------------------------------
extern "C" void kernel_launch(void* const* d_in, const int* in_sizes, int n_in,
                              void* d_out, int out_size, void* d_ws, size_t ws_size,
                              hipStream_t stream) {
  const float* x  = (const float*)d_in[0];   // (P,1)
  const float* y  = (const float*)d_in[1];   // (P,1)
  const float* F  = (const float*)d_in[2];   // (2,128,129,129)
  const float* Wc = (const float*)d_in[3];   // (128,1,3,3)
  float* out = (float*)d_out;                // (P,128)
  float* A   = (float*)d_ws;                 // alpha transposed: (2,129,129,128) = 17 MB

  const int P = in_sizes[0];

  // Stage 1: conv + tanh into transposed workspace (fully overwritten each call)
  const int conv_blocks = NCELL * HH * NXCHUNK;      // 2*129*9 = 2322
  conv_tanh_tr_kernel<<<conv_blocks, 128, 0, stream>>>(F, Wc, A);

  // Stage 2: 32 points per 256-thread block
  const int blocks = (P + PTS_PER_BLOCK - 1) / PTS_PER_BLOCK;
  sample_cosine_kernel<<<blocks, 256, 0, stream>>>(x, y, A, out, P);
}